// GCN_22995254903253
// MI455X (gfx1250) — compile-verified
//
#include <hip/hip_runtime.h>

#define D 64

typedef __attribute__((ext_vector_type(2))) float v2f;
typedef __attribute__((ext_vector_type(8))) float v8f;

// ---------------- degree / normalization ----------------
__global__ void k_deg_init(float* __restrict__ deg, int n) {
    int i = blockIdx.x * blockDim.x + threadIdx.x;
    if (i < n) deg[i] = 1.0f;                      // self-loop contributes 1
}

__global__ void k_deg_accum(const int* __restrict__ dst, float* __restrict__ deg, int e) {
    int i = blockIdx.x * blockDim.x + threadIdx.x;
    if (i < e) atomicAdd(&deg[dst[i]], 1.0f);
}

__global__ void k_dinv(const float* __restrict__ deg, float* __restrict__ dinv, int n) {
    int i = blockIdx.x * blockDim.x + threadIdx.x;
    if (i < n) {
        float d = deg[i];
        dinv[i] = (d > 0.0f) ? rsqrtf(d) : 0.0f;
    }
}

// ---------------- GEMM: Y[n x 64] = X[n x 64] @ W[64 x 64] ----------------
// One wave computes one 16x16 output tile via v_wmma_f32_16x16x4_f32.
// Block = 256 threads = 8 waves = 2 row-tiles x 4 col-tiles.
__global__ void __launch_bounds__(256)
k_gemm_wmma(const float* __restrict__ X, const float* __restrict__ W,
            float* __restrict__ Y, int n) {
    const int lane = threadIdx.x & 31;
    const int wave = threadIdx.x >> 5;
    const int rowTiles = (n + 15) >> 4;
    const int rt = blockIdx.x * 2 + (wave >> 2);
    const int ct = wave & 3;
    if (rt >= rowTiles) return;                    // uniform per-wave branch

    const int r0 = rt << 4;
    const int c0 = ct << 4;
    const int m  = lane & 15;                      // A: row-in-tile, B: col-in-tile
    const int kh = lane >> 4;                      // K half (0 -> K{0,1}, 1 -> K{2,3})

    int arow = r0 + m;
    if (arow >= n) arow = n - 1;                   // clamp (no EXEC divergence)
    const float* __restrict__ aptr = X + (size_t)arow * D;
    const float* __restrict__ bptr = W + c0 + m;   // column c0+m of W (row-major 64x64)

    v8f acc = {};
#pragma unroll
    for (int k0 = 0; k0 < D; k0 += 4) {
        // A fragment 16x4: lane holds X[r0+m][k0 + 2*kh + {0,1}]
        v2f a = *(const v2f*)(aptr + k0 + 2 * kh);
        // B fragment 4x16: lane holds W[k0 + 2*kh + {0,1}][c0+m]
        v2f b;
        b.x = bptr[(k0 + 2 * kh)     * D];
        b.y = bptr[(k0 + 2 * kh + 1) * D];
        acc = __builtin_amdgcn_wmma_f32_16x16x4_f32(
            /*neg_a=*/false, a, /*neg_b=*/false, b,
            /*c_mod=*/(short)0, acc, /*reuse_a=*/false, /*reuse_b=*/false);
    }

    // C/D layout: VGPR r -> row (r0 + r + 8*kh), col (c0 + m)
    const int col = c0 + m;
#pragma unroll
    for (int r = 0; r < 8; ++r) {
        int row = r0 + r + 8 * kh;
        if (row < n) Y[(size_t)row * D + col] = acc[r];
    }
}

// ---------------- self-loop + bias init: agg[v][d] = dinv[v]^2 * xw[v][d] + b[d] ----
__global__ void k_self_bias(const float* __restrict__ xw, const float* __restrict__ dinv,
                            const float* __restrict__ b, float* __restrict__ agg, int total) {
    int i = blockIdx.x * blockDim.x + threadIdx.x;
    if (i < total) {
        int v = i >> 6;
        int d = i & 63;
        float s = dinv[v];
        agg[i] = s * s * xw[i] + b[d];
    }
}

// ---------------- edge scatter: agg[dst] += dinv[src]*dinv[dst]*xw[src] -------------
// One wave per edge; each lane handles features {lane, lane+32}. Atomics land in L2
// (12.8 MB agg buffer << 192 MB L2).
__global__ void __launch_bounds__(256)
k_scatter(const int* __restrict__ src, const int* __restrict__ dst,
          const float* __restrict__ dinv, const float* __restrict__ xw,
          float* __restrict__ agg, int e) {
    int edge = (blockIdx.x << 3) + (threadIdx.x >> 5);
    if (edge >= e) return;
    int s = src[edge];
    int t = dst[edge];
    float nrm = dinv[s] * dinv[t];
    int lane = threadIdx.x & 31;
    const float* __restrict__ xs = xw + (size_t)s * D;
    float* __restrict__ at = agg + (size_t)t * D;
    atomicAdd(at + lane,      nrm * xs[lane]);
    atomicAdd(at + lane + 32, nrm * xs[lane + 32]);
}

// ---------------- ReLU in place ----------------
__global__ void k_relu(float* __restrict__ h, int total) {
    int i = blockIdx.x * blockDim.x + threadIdx.x;
    if (i < total) h[i] = fmaxf(h[i], 0.0f);
}

// ---------------- driver ----------------
extern "C" void kernel_launch(void* const* d_in, const int* in_sizes, int n_in,
                              void* d_out, int out_size, void* d_ws, size_t ws_size,
                              hipStream_t stream) {
    const float* x  = (const float*)d_in[0];
    const int*   ei = (const int*)  d_in[1];
    const float* W1 = (const float*)d_in[2];
    const float* b1 = (const float*)d_in[3];
    const float* W2 = (const float*)d_in[4];
    const float* b2 = (const float*)d_in[5];

    const int n = in_sizes[0] / D;
    const int e = in_sizes[1] / 2;
    const int* src = ei;
    const int* dst = ei + e;

    float* ws   = (float*)d_ws;
    float* deg  = ws;                         // n floats
    float* dinv = ws + n;                     // n floats
    float* xw   = ws + 2 * (size_t)n;         // n*64 floats
    float* agg  = xw + (size_t)n * D;         // n*64 floats
    float* out  = (float*)d_out;              // n*64 floats

    const int T = 256;
    const int total = n * D;
    const int rowTiles = (n + 15) / 16;
    const int gemmBlocks = (rowTiles + 1) / 2;
    const int scatBlocks = (e + 7) / 8;

    // graph normalization (recomputed every call: deterministic, capture-safe)
    k_deg_init <<<(n + T - 1) / T, T, 0, stream>>>(deg, n);
    k_deg_accum<<<(e + T - 1) / T, T, 0, stream>>>(dst, deg, e);
    k_dinv     <<<(n + T - 1) / T, T, 0, stream>>>(deg, dinv, n);

    // layer 1: h = relu( Dhat (x@W1) + b1 )
    k_gemm_wmma<<<gemmBlocks, 256, 0, stream>>>(x, W1, xw, n);
    k_self_bias<<<(total + T - 1) / T, T, 0, stream>>>(xw, dinv, b1, agg, total);
    k_scatter  <<<scatBlocks, 256, 0, stream>>>(src, dst, dinv, xw, agg, e);
    k_relu     <<<(total + T - 1) / T, T, 0, stream>>>(agg, total);

    // layer 2: out = Dhat (h@W2) + b2
    k_gemm_wmma<<<gemmBlocks, 256, 0, stream>>>(agg, W2, xw, n);
    k_self_bias<<<(total + T - 1) / T, T, 0, stream>>>(xw, dinv, b2, out, total);
    k_scatter  <<<scatBlocks, 256, 0, stream>>>(src, dst, dinv, xw, out, e);
}